// COVPool_36447092474532
// MI455X (gfx1250) — compile-verified
//
#include <hip/hip_runtime.h>
#include <math.h>

// ---------------------------------------------------------------------------
// COVPool (iSQRT-COV) for MI455X / gfx1250, fp32 WMMA path.
//
// Shapes: x[16,256,28,28], raw_attentions[16,16,28,28].
// Key identity: with z[c,(a,hw)] = A[a,hw]*x[c,hw],
//   sum_m z_c z_d = sum_hw S[hw] x_c x_d,  S = sum_a A^2
//   sum_m z_c     = sum_hw T[hw] x_c,      T = sum_a A
// -> covariance GEMM has K = 784 instead of 12544 (26 GFLOP -> 1.6 GFLOP),
//    and z (205 MB) is never materialized.
// All matrix math in fp32 via V_WMMA_F32_16X16X4_F32 (wave32, 2x2 tile/wave).
// Newton-Schulz elementwise steps (0.5*(3I - P)), final sqrt(trace) scale and
// triu extraction are fused into the GEMM epilogues.
// ---------------------------------------------------------------------------

typedef float v2f __attribute__((ext_vector_type(2)));
typedef float v8f __attribute__((ext_vector_type(8)));

#define B_    16
#define C_    256
#define HW    784
#define M_ATN 16
#define M_TOT (M_ATN * HW)           // 12544
#define CC    (C_ * C_)              // 65536
#define TRIU  ((C_ * (C_ + 1)) / 2)  // 32896
#define ITERN 5

#define GEMM_PLAIN 0   // D = A@B
#define GEMM_ZY    1   // D = 0.5*(3I - A@B)
#define GEMM_TRIU  2   // out_triu = alpha_b * (A@B), upper triangle only

// ---- attentions = sigmoid(raw)+eps ; S = sum_a att^2 ; T = sum_a att -------
__global__ void attn_stats_kernel(const float* __restrict__ raw,
                                  float* __restrict__ att_out,
                                  float* __restrict__ S,
                                  float* __restrict__ T) {
    int idx = blockIdx.x * blockDim.x + threadIdx.x;   // b*HW + hw
    if (idx >= B_ * HW) return;
    int b = idx / HW, hw = idx % HW;
    const float* rp = raw + (size_t)b * M_TOT + hw;
    float* ap = att_out + (size_t)b * M_TOT + hw;
    float s = 0.f, t = 0.f;
    #pragma unroll
    for (int a = 0; a < M_ATN; ++a) {
        float v = 1.0f / (1.0f + expf(-rp[a * HW])) + 1e-12f;
        ap[a * HW] = v;
        s += v * v;
        t += v;
    }
    S[idx] = s;
    T[idx] = t;
}

// ---- tvec[b,c] = sum_hw T[b,hw] * x[b,c,hw] --------------------------------
__global__ void tvec_kernel(const float* __restrict__ x,
                            const float* __restrict__ T,
                            float* __restrict__ tvec) {
    int idx = blockIdx.x * blockDim.x + threadIdx.x;   // b*C + c
    if (idx >= B_ * C_) return;
    int b = idx / C_;
    const float* xp = x + (size_t)idx * HW;
    const float* Tp = T + (size_t)b * HW;
    float acc = 0.f;
    for (int i = 0; i < HW; ++i) acc += Tp[i] * xp[i];
    tvec[idx] = acc;
}

// ---- G[b] = (S-weighted X) @ X^T, X = x[b] (256 x 784), WMMA fp32 ----------
// One wave -> 32x32 output (2x2 of 16x16). 64 wave-tiles/batch, 1024 waves.
__global__ void gram_wmma_kernel(const float* __restrict__ x,
                                 const float* __restrict__ S,
                                 float* __restrict__ G) {
    int wave = (blockIdx.x * blockDim.x + threadIdx.x) >> 5;  // 0..1023
    int lane = threadIdx.x & 31;
    int b    = wave >> 6;
    int tid  = wave & 63;
    int r0   = (tid >> 3) * 32;
    int c0   = (tid & 7) * 32;
    int half = lane >> 4;
    int lr   = lane & 15;

    const float* xb  = x + (size_t)b * C_ * HW;
    const float* Sb  = S + (size_t)b * HW;
    const float* Ar0 = xb + (size_t)(r0 + lr) * HW;
    const float* Ar1 = xb + (size_t)(r0 + 16 + lr) * HW;
    const float* Bc0 = xb + (size_t)(c0 + lr) * HW;
    const float* Bc1 = xb + (size_t)(c0 + 16 + lr) * HW;

    v8f acc00 = {}, acc01 = {}, acc10 = {}, acc11 = {};
    for (int k = 0; k < HW; k += 4) {
        int kk = k + 2 * half;                 // A: lanes<16 -> K 0,1 ; lanes>=16 -> K 2,3
        float s0 = Sb[kk], s1 = Sb[kk + 1];
        v2f a0 = { s0 * Ar0[kk], s1 * Ar0[kk + 1] };
        v2f a1 = { s0 * Ar1[kk], s1 * Ar1[kk + 1] };
        v2f b0 = { Bc0[kk], Bc0[kk + 1] };     // B[K][N] = x[col][hw]
        v2f b1 = { Bc1[kk], Bc1[kk + 1] };
        acc00 = __builtin_amdgcn_wmma_f32_16x16x4_f32(false, a0, false, b0, (short)0, acc00, false, false);
        acc01 = __builtin_amdgcn_wmma_f32_16x16x4_f32(false, a0, false, b1, (short)0, acc01, false, false);
        acc10 = __builtin_amdgcn_wmma_f32_16x16x4_f32(false, a1, false, b0, (short)0, acc10, false, false);
        acc11 = __builtin_amdgcn_wmma_f32_16x16x4_f32(false, a1, false, b1, (short)0, acc11, false, false);
    }

    float* Gb = G + (size_t)b * CC;
    #pragma unroll
    for (int v = 0; v < 8; ++v) {
        int row = r0 + v + 8 * half;           // C/D layout: VGPR v -> M = v (+8 for high half)
        Gb[(size_t)row * C_ + c0 + lr]             = acc00[v];
        Gb[(size_t)row * C_ + c0 + 16 + lr]        = acc01[v];
        Gb[(size_t)(row + 16) * C_ + c0 + lr]      = acc10[v];
        Gb[(size_t)(row + 16) * C_ + c0 + 16 + lr] = acc11[v];
    }
}

// ---- per-batch trace of cov and sqrt for final rescale ---------------------
__global__ void trace_kernel(const float* __restrict__ G,
                             const float* __restrict__ tvec,
                             float* __restrict__ normA,
                             float* __restrict__ salpha) {
    __shared__ float sg[256];
    __shared__ float st[256];
    int b = blockIdx.x, c = threadIdx.x;
    sg[c] = G[(size_t)b * CC + (size_t)c * C_ + c];
    float tv = tvec[b * C_ + c];
    st[c] = tv * tv;
    __syncthreads();
    for (int s = 128; s > 0; s >>= 1) {
        if (c < s) { sg[c] += sg[c + s]; st[c] += st[c + s]; }
        __syncthreads();
    }
    if (c == 0) {
        const float invM = 1.0f / (float)M_TOT;
        float tr = sg[0] * invM - st[0] * invM * invM;   // trace(cov)
        normA[b]  = tr;
        salpha[b] = sqrtf(tr);
    }
}

// ---- An = (G/M - t t^T / M^2)/normA in place on G; Z = 0.5*(3I - An) -------
__global__ void an_zy_kernel(float* __restrict__ G,
                             const float* __restrict__ tvec,
                             const float* __restrict__ normA,
                             float* __restrict__ Z) {
    int idx = blockIdx.x * blockDim.x + threadIdx.x;
    if (idx >= B_ * CC) return;
    int b = idx / CC, rc = idx % CC;
    int r = rc / C_, c = rc % C_;
    const float invM = 1.0f / (float)M_TOT;
    float v = G[idx] * invM - tvec[b * C_ + r] * tvec[b * C_ + c] * invM * invM;
    v = v / normA[b];
    G[idx] = v;
    Z[idx] = 0.5f * (((r == c) ? 3.0f : 0.0f) - v);
}

// ---- fused epilogue store helper -------------------------------------------
__device__ __forceinline__ void gemm_store(float* __restrict__ D, int b,
                                           int row, int col, float val,
                                           float al, int mode) {
    if (mode == GEMM_PLAIN) {
        D[(size_t)b * CC + (size_t)row * C_ + col] = val;
    } else if (mode == GEMM_ZY) {
        D[(size_t)b * CC + (size_t)row * C_ + col] =
            0.5f * (((row == col) ? 3.0f : 0.0f) - val);
    } else {  // GEMM_TRIU: row-major upper triangle incl. diagonal
        if (col >= row) {
            int base = row * C_ - (row * (row - 1)) / 2;
            D[(size_t)b * TRIU + base + (col - row)] = al * val;
        }
    }
}

// ---- batched 256x256x256 fp32 WMMA GEMM with fused epilogues ---------------
__global__ void gemm_wmma_kernel(const float* __restrict__ A,
                                 const float* __restrict__ Bm,
                                 float* __restrict__ D,
                                 const float* __restrict__ alpha,
                                 int mode) {
    int wave = (blockIdx.x * blockDim.x + threadIdx.x) >> 5;  // 0..1023
    int lane = threadIdx.x & 31;
    int b    = wave >> 6;
    int tid  = wave & 63;
    int r0   = (tid >> 3) * 32;
    int c0   = (tid & 7) * 32;
    int half = lane >> 4;
    int lr   = lane & 15;

    const float* Ab  = A  + (size_t)b * CC;
    const float* Bb  = Bm + (size_t)b * CC;
    const float* Ar0 = Ab + (size_t)(r0 + lr) * C_;
    const float* Ar1 = Ab + (size_t)(r0 + 16 + lr) * C_;

    v8f acc00 = {}, acc01 = {}, acc10 = {}, acc11 = {};
    for (int k = 0; k < C_; k += 4) {
        int kk = k + 2 * half;
        v2f a0 = { Ar0[kk], Ar0[kk + 1] };
        v2f a1 = { Ar1[kk], Ar1[kk + 1] };
        const float* Bp = Bb + (size_t)kk * C_;
        v2f b0 = { Bp[c0 + lr],      Bp[C_ + c0 + lr] };
        v2f b1 = { Bp[c0 + 16 + lr], Bp[C_ + c0 + 16 + lr] };
        acc00 = __builtin_amdgcn_wmma_f32_16x16x4_f32(false, a0, false, b0, (short)0, acc00, false, false);
        acc01 = __builtin_amdgcn_wmma_f32_16x16x4_f32(false, a0, false, b1, (short)0, acc01, false, false);
        acc10 = __builtin_amdgcn_wmma_f32_16x16x4_f32(false, a1, false, b0, (short)0, acc10, false, false);
        acc11 = __builtin_amdgcn_wmma_f32_16x16x4_f32(false, a1, false, b1, (short)0, acc11, false, false);
    }

    float al = (alpha != nullptr) ? alpha[b] : 1.0f;
    #pragma unroll
    for (int v = 0; v < 8; ++v) {
        int row = r0 + v + 8 * half;
        gemm_store(D, b, row,      c0 + lr,      acc00[v], al, mode);
        gemm_store(D, b, row,      c0 + 16 + lr, acc01[v], al, mode);
        gemm_store(D, b, row + 16, c0 + lr,      acc10[v], al, mode);
        gemm_store(D, b, row + 16, c0 + 16 + lr, acc11[v], al, mode);
    }
}

// ---------------------------------------------------------------------------
extern "C" void kernel_launch(void* const* d_in, const int* in_sizes, int n_in,
                              void* d_out, int out_size, void* d_ws, size_t ws_size,
                              hipStream_t stream) {
    (void)in_sizes; (void)n_in; (void)out_size; (void)ws_size;
    const float* x   = (const float*)d_in[0];
    const float* raw = (const float*)d_in[1];
    float* out     = (float*)d_out;            // [B, 32896] triu output
    float* att_out = out + (size_t)B_ * TRIU;  // [B, 16, 28, 28] attentions output

    float* ws = (float*)d_ws;
    size_t off = 0;
    float* S      = ws + off; off += (size_t)B_ * HW;
    float* T      = ws + off; off += (size_t)B_ * HW;
    float* tv     = ws + off; off += (size_t)B_ * C_;
    float* normA  = ws + off; off += 16;
    float* salpha = ws + off; off += 16;
    float* G      = ws + off; off += (size_t)B_ * CC;  // cov -> An (in place)
    float* Y      = ws + off; off += (size_t)B_ * CC;
    float* Y2     = ws + off; off += (size_t)B_ * CC;
    float* Z      = ws + off; off += (size_t)B_ * CC;
    float* Z2     = ws + off; off += (size_t)B_ * CC;
    float* P      = ws + off; off += (size_t)B_ * CC;
    // total ws use: ~25.3 MB of fp32 (all matrices L2-resident)

    const int elt_blocks = (B_ * CC) / 256;  // 4096

    attn_stats_kernel<<<(B_ * HW + 255) / 256, 256, 0, stream>>>(raw, att_out, S, T);
    tvec_kernel<<<(B_ * C_ + 255) / 256, 256, 0, stream>>>(x, T, tv);
    gram_wmma_kernel<<<128, 256, 0, stream>>>(x, S, G);
    trace_kernel<<<B_, 256, 0, stream>>>(G, tv, normA, salpha);
    an_zy_kernel<<<elt_blocks, 256, 0, stream>>>(G, tv, normA, Z);  // G := An, Z := ZY0

    // Y = An @ ZY0
    gemm_wmma_kernel<<<128, 256, 0, stream>>>(G, Z, Y, nullptr, GEMM_PLAIN);

    float* Yc = Y;  float* Ya = Y2;
    float* Zc = Z;  float* Za = Z2;
    for (int it = 1; it < ITERN - 1; ++it) {   // 3 iterations for ITERN=5
        gemm_wmma_kernel<<<128, 256, 0, stream>>>(Zc, Yc, P, nullptr, GEMM_ZY);     // P = 0.5*(3I - Z@Y)
        gemm_wmma_kernel<<<128, 256, 0, stream>>>(Yc, P, Ya, nullptr, GEMM_PLAIN);  // Ynew = Y@P
        gemm_wmma_kernel<<<128, 256, 0, stream>>>(P, Zc, Za, nullptr, GEMM_PLAIN);  // Znew = P@Z
        float* t0 = Yc; Yc = Ya; Ya = t0;
        float* t1 = Zc; Zc = Za; Za = t1;
    }
    // P = 0.5*(3I - Z@Y) ; out_triu = sqrt(trace) * (Y @ P)
    gemm_wmma_kernel<<<128, 256, 0, stream>>>(Zc, Yc, P, nullptr, GEMM_ZY);
    gemm_wmma_kernel<<<128, 256, 0, stream>>>(Yc, P, out, salpha, GEMM_TRIU);
}